// cross_set_score_15564961480815
// MI455X (gfx1250) — compile-verified
//
#include <hip/hip_runtime.h>
#include <hip/hip_bf16.h>

typedef __attribute__((ext_vector_type(2))) float v2f;
typedef __attribute__((ext_vector_type(8))) float v8f;

#define D_IN      512
#define NSET      32
#define NITEM     128
#define NHEAD     8
#define HEADSZ    64
#define MROWS     (NSET * NITEM)   // 4096 rows per tensor

// ---------------------------------------------------------------------------
// Kernel 1: projection  OUT[4096,512] = IN[4096,512] @ W1[512,512]
// One wave per 32x32 macro-tile (2x2 of 16x16), K=512 via 128 steps of
// v_wmma_f32_16x16x4_f32 per 16x16 tile. A fragments shared across the two
// N-tiles, B fragments shared across the two M-tiles -> 8 B/lane per WMMA.
// Macro-tiles per tensor: (4096/32)*(512/32) = 128*16 = 2048 -> 256 blocks
// of 8 waves. blockIdx.y: 0 -> x tensor, 1 -> y tensor.
// ---------------------------------------------------------------------------
__global__ __launch_bounds__(256) void proj_kernel(const float* __restrict__ X,
                                                   const float* __restrict__ Y,
                                                   const float* __restrict__ W1,
                                                   float* __restrict__ XP,
                                                   float* __restrict__ YP) {
    const int wave = threadIdx.x >> 5;
    const int lane = threadIdx.x & 31;
    const int lr   = lane & 15;      // M (A) / N (B,C) index within tile
    const int hi   = lane >> 4;      // K half-select (A/B), M+8 select (C/D)

    const int t   = blockIdx.x * 8 + wave;       // 0..2047
    const int tn2 = t & 15;                      // 16 macro-tiles along N=512
    const int tm2 = t >> 4;                      // 128 macro-tiles along M=4096
    const int m0  = tm2 * 32;
    const int n0  = tn2 * 32;

    const float* IN  = blockIdx.y ? Y  : X;
    float*       OUT = blockIdx.y ? YP : XP;

    const float* arow0 = IN + (size_t)(m0 + lr)      * D_IN + 2 * hi;
    const float* arow1 = IN + (size_t)(m0 + 16 + lr) * D_IN + 2 * hi;
    const float* bcol0 = W1 + n0 + lr;
    const float* bcol1 = W1 + n0 + 16 + lr;

    v8f c00 = {}, c01 = {}, c10 = {}, c11 = {};
#pragma unroll 4
    for (int kk = 0; kk < D_IN; kk += 4) {
        const int k0 = kk + 2 * hi;
        v2f a0 = *(const v2f*)(arow0 + kk);                 // one b64 each
        v2f a1 = *(const v2f*)(arow1 + kk);
        v2f b0, b1;                                         // two b32 each
        b0.x = bcol0[(size_t)k0 * D_IN];
        b0.y = bcol0[(size_t)(k0 + 1) * D_IN];
        b1.x = bcol1[(size_t)k0 * D_IN];
        b1.y = bcol1[(size_t)(k0 + 1) * D_IN];
        c00 = __builtin_amdgcn_wmma_f32_16x16x4_f32(false, a0, false, b0, (short)0, c00, false, false);
        c01 = __builtin_amdgcn_wmma_f32_16x16x4_f32(false, a0, false, b1, (short)0, c01, false, false);
        c10 = __builtin_amdgcn_wmma_f32_16x16x4_f32(false, a1, false, b0, (short)0, c10, false, false);
        c11 = __builtin_amdgcn_wmma_f32_16x16x4_f32(false, a1, false, b1, (short)0, c11, false, false);
    }

    // C/D layout: vgpr r -> row (r + 8*hi), col lr
    float* o00 = OUT + (size_t)(m0 + 8 * hi)      * D_IN + n0 + lr;
    float* o10 = OUT + (size_t)(m0 + 16 + 8 * hi) * D_IN + n0 + lr;
#pragma unroll
    for (int r = 0; r < 8; ++r) {
        o00[(size_t)r * D_IN]      = c00[r];
        o00[(size_t)r * D_IN + 16] = c01[r];
        o10[(size_t)r * D_IN]      = c10[r];
        o10[(size_t)r * D_IN + 16] = c11[r];
    }
}

// ---------------------------------------------------------------------------
// Kernel 2: per-pair scores + LeakyReLU + pooling + head mix.
// grid = 1024 blocks (pair a,b); block = 256 threads = 8 waves (wave = head).
// Each wave: 128x128 score matrix as 2(k) x 4(m) register-tiled groups of
// 16x16 tiles, K=64 (16 WMMA steps). 8 independent accumulator chains; A
// fragments shared across 4 m-tiles, B fragments across 2 k-tiles ->
// 6 x b64 loads per 8 WMMAs. xp/yp layout [a,k,h,l]: element at
// (a*128+k)*512 + h*64 + l, so every fragment is one aligned b64 per lane.
// ---------------------------------------------------------------------------
__global__ __launch_bounds__(256) void score_kernel(const float* __restrict__ XP,
                                                    const float* __restrict__ YP,
                                                    const int* __restrict__ nItem,
                                                    const float* __restrict__ W2,
                                                    float* __restrict__ out) {
    __shared__ float pooled[NHEAD];

    const int h    = threadIdx.x >> 5;   // head = wave id
    const int lane = threadIdx.x & 31;
    const int lr   = lane & 15;
    const int hi   = lane >> 4;

    const int p = blockIdx.x;            // pair index
    const int a = p >> 5;
    const int b = p & 31;

    const float* Xbase = XP + (size_t)(a * NITEM + lr) * D_IN + h * HEADSZ + 2 * hi;
    const float* Ybase = YP + (size_t)(b * NITEM + lr) * D_IN + h * HEADSZ + 2 * hi;

    float acc = 0.f;
    for (int tk = 0; tk < 8; tk += 2) {
        const float* x0 = Xbase + (size_t)tk * 16 * D_IN;
        const float* x1 = x0 + (size_t)16 * D_IN;
        for (int tm = 0; tm < 8; tm += 4) {
            const float* y0 = Ybase + (size_t)tm * 16 * D_IN;
            const float* y1 = y0 + (size_t)16 * D_IN;
            const float* y2 = y1 + (size_t)16 * D_IN;
            const float* y3 = y2 + (size_t)16 * D_IN;
            v8f c00 = {}, c01 = {}, c02 = {}, c03 = {};
            v8f c10 = {}, c11 = {}, c12 = {}, c13 = {};
#pragma unroll
            for (int kk = 0; kk < HEADSZ; kk += 4) {
                v2f a0 = *(const v2f*)(x0 + kk);
                v2f a1 = *(const v2f*)(x1 + kk);
                v2f b0 = *(const v2f*)(y0 + kk);
                v2f b1 = *(const v2f*)(y1 + kk);
                v2f b2 = *(const v2f*)(y2 + kk);
                v2f b3 = *(const v2f*)(y3 + kk);
                c00 = __builtin_amdgcn_wmma_f32_16x16x4_f32(false, a0, false, b0, (short)0, c00, false, false);
                c01 = __builtin_amdgcn_wmma_f32_16x16x4_f32(false, a0, false, b1, (short)0, c01, false, false);
                c02 = __builtin_amdgcn_wmma_f32_16x16x4_f32(false, a0, false, b2, (short)0, c02, false, false);
                c03 = __builtin_amdgcn_wmma_f32_16x16x4_f32(false, a0, false, b3, (short)0, c03, false, false);
                c10 = __builtin_amdgcn_wmma_f32_16x16x4_f32(false, a1, false, b0, (short)0, c10, false, false);
                c11 = __builtin_amdgcn_wmma_f32_16x16x4_f32(false, a1, false, b1, (short)0, c11, false, false);
                c12 = __builtin_amdgcn_wmma_f32_16x16x4_f32(false, a1, false, b2, (short)0, c12, false, false);
                c13 = __builtin_amdgcn_wmma_f32_16x16x4_f32(false, a1, false, b3, (short)0, c13, false, false);
            }
            // LeakyReLU(0.3) then accumulate (the /sqrt(64) scale commutes)
#pragma unroll
            for (int r = 0; r < 8; ++r) {
                float v;
                v = c00[r]; acc += (v >= 0.f) ? v : 0.3f * v;
                v = c01[r]; acc += (v >= 0.f) ? v : 0.3f * v;
                v = c02[r]; acc += (v >= 0.f) ? v : 0.3f * v;
                v = c03[r]; acc += (v >= 0.f) ? v : 0.3f * v;
                v = c10[r]; acc += (v >= 0.f) ? v : 0.3f * v;
                v = c11[r]; acc += (v >= 0.f) ? v : 0.3f * v;
                v = c12[r]; acc += (v >= 0.f) ? v : 0.3f * v;
                v = c13[r]; acc += (v >= 0.f) ? v : 0.3f * v;
            }
        }
    }
    acc *= 0.125f;  // 1/sqrt(HEAD_SIZE)

    // wave32 reduction
#pragma unroll
    for (int off = 16; off > 0; off >>= 1)
        acc += __shfl_xor(acc, off, 32);
    if (lane == 0) pooled[h] = acc;
    __syncthreads();

    if (threadIdx.x == 0) {
        float r = 0.f;
#pragma unroll
        for (int hh = 0; hh < NHEAD; ++hh) r += pooled[hh] * W2[hh];
        out[p] = r / ((float)nItem[b] * (float)NITEM);
    }
}

// ---------------------------------------------------------------------------
extern "C" void kernel_launch(void* const* d_in, const int* in_sizes, int n_in,
                              void* d_out, int out_size, void* d_ws, size_t ws_size,
                              hipStream_t stream) {
    const float* x     = (const float*)d_in[0];  // [32,128,512]
    const float* y     = (const float*)d_in[1];  // [32,128,512]
    const int*   nIt   = (const int*)  d_in[2];  // [32]
    const float* W1    = (const float*)d_in[3];  // [512,512]
    const float* W2    = (const float*)d_in[4];  // [8,1]
    float*       out   = (float*)d_out;          // [32,32,1]

    float* xp = (float*)d_ws;                         // 4096*512 floats = 8 MB
    float* yp = xp + (size_t)MROWS * D_IN;            // next 8 MB

    proj_kernel<<<dim3(256, 2), 256, 0, stream>>>(x, y, W1, xp, yp);
    score_kernel<<<dim3(NSET * NSET), 256, 0, stream>>>(xp, yp, nIt, W2, out);
}